// RiskSensitiveGNN_86723979640908
// MI455X (gfx1250) — compile-verified
//
#include <hip/hip_runtime.h>
#include <math.h>

#define N_NODES    50000
#define N_EDGES    1600000
#define N_EDGES_T  (N_EDGES + N_NODES)   // with self loops
#define BATCHES    64
#define NODE_DIM   8
#define EDGE_DIM   4
#define HID        128
#define OUT_DIM    256
#define LN_EPS     1e-5f
#define NEG_SLOPE  0.2f
#define BK         32                    // K-chunk staged in LDS per iteration

typedef unsigned int       u32;
typedef unsigned long long u64;
typedef __attribute__((ext_vector_type(2))) float        v2f;
typedef __attribute__((ext_vector_type(8))) float        v8f;
typedef __attribute__((ext_vector_type(4))) unsigned int v4u;
typedef __attribute__((ext_vector_type(8))) int          v8i;
typedef __attribute__((ext_vector_type(4))) int          v4i;

__device__ __forceinline__ float waveReduceSum(float v) {
#pragma unroll
  for (int m = 16; m >= 1; m >>= 1) v += __shfl_xor(v, m, 32);
  return v;
}

__device__ __forceinline__ void atomicMaxF(float* addr, float val) {
  unsigned int* a = (unsigned int*)addr;
  unsigned int old = *a;
  while (true) {
    float f = __uint_as_float(old);
    if (f >= val) break;
    unsigned int assumed = old;
    old = atomicCAS(a, assumed, __float_as_uint(val));
    if (old == assumed) break;
  }
}

// ---------------------------------------------------------------------------
// TDM 2D tile load: global (tensor_d1 rows x tensor_d0 cols of f32, row stride
// d0_stride) -> LDS at byte offset lds_off, tile = tile_d1 rows x tile_d0 cols.
// D# bitfields per CDNA5 ISA 8.3/8.4: group0 {count=1, lds_addr, global_addr,
// type=2}, group1 {data_size=4B, tensor dims, tile dims, dim0 stride}.
// OOB rows/cols read as zero (handles the ragged last row-block for free).
// 6-arg builtin form (clang-23 / amdgpu-toolchain): groups 2/3 + extra octet
// zeroed (2D tensor), cpol=0.
// ---------------------------------------------------------------------------
__device__ __forceinline__ void tdm_load_2d(u32 lds_off, const float* gptr,
                                            u32 tensor_d0, u32 tensor_d1,
                                            u32 tile_d0, u32 tile_d1,
                                            u32 d0_stride) {
  u64 ga = (u64)(uintptr_t)gptr;
  v4u g0;
  g0.x = 0x1u;                                            // count=1 (valid user D#)
  g0.y = lds_off;                                         // lds_addr (bytes)
  g0.z = (u32)ga;                                         // global_addr[31:0]
  g0.w = ((u32)(ga >> 32) & 0x01FFFFFFu) | (2u << 30);    // addr[56:32] | type=2
  v8i g1;
  g1[0] = (int)(2u << 16);                                // data_size=4B, wg_mask=0
  g1[1] = (int)((tensor_d0 & 0xFFFFu) << 16);             // tensor_dim0[15:0]
  g1[2] = (int)(((tensor_d0 >> 16) & 0xFFFFu) |
                ((tensor_d1 & 0xFFFFu) << 16));           // dim0 hi | dim1 lo
  g1[3] = (int)(((tensor_d1 >> 16) & 0xFFFFu) |
                ((tile_d0 & 0xFFFFu) << 16));             // dim1 hi | tile_dim0
  g1[4] = (int)(tile_d1 & 0xFFFFu);                       // tile_dim1 (tile_dim2=0)
  g1[5] = (int)d0_stride;                                 // tensor_dim0_stride lo
  g1[6] = 0;                                              // stride hi / dim1_stride
  g1[7] = 0;
  v4i z4 = {0, 0, 0, 0};
  v8i z8 = {0, 0, 0, 0, 0, 0, 0, 0};
  __builtin_amdgcn_tensor_load_to_lds(g0, g1, z4, z4, z8, 0);
}

// ---------------------------------------------------------------------------
// xt[N,128] = A[N,K] @ W[K,128] via V_WMMA_F32_16X16X4_F32.
// 256-thread block (8 waves): wave w owns row tile (blockIdx*8+w) and all 8
// column tiles (acc[8]).  Per BK-chunk, wave 0 issues two TDM loads
// (A 128xBK, W BKx128) into LDS, waits TENSORcnt, block barriers, then the
// inner loop is ds_load + v_wmma only.
// A-frag (16x4 f32): lanes 0-15 -> M=lane, K=0,1 ; lanes 16-31 -> M=lane-16, K=2,3
// B-frag (4x16 f32): lanes 0-15 -> N=lane, K=0,1 ; lanes 16-31 -> N=lane-16, K=2,3
// C/D (16x16 f32):   vgpr r: lanes 0-15 -> M=r, lanes 16-31 -> M=r+8, N=lane%16
// ---------------------------------------------------------------------------
__global__ __launch_bounds__(256) void gemm_xt_wmma(const float* __restrict__ A, int lda, int K,
                                                    const float* __restrict__ W,
                                                    float* __restrict__ xt) {
  __shared__ float smem[128 * BK + BK * HID];   // sA (128 x kc) | sW (kc x 128)
  float* sA = smem;
  float* sW = smem + 128 * BK;
  const int lane     = threadIdx.x & 31;
  const int wave     = threadIdx.x >> 5;
  const int tm       = blockIdx.x * 8 + wave;   // this wave's row tile
  const int nRowTiles = N_NODES / 16;           // 3125
  const int row0     = blockIdx.x * 128;        // block row base
  const int khalf    = (lane >> 4) << 1;        // 0 | 2
  const int lrow     = wave * 16 + (lane & 15); // local A row for A-frag
  const int ncb      = lane & 15;               // column within a 16-wide tile

  v8f acc[8];
#pragma unroll
  for (int t = 0; t < 8; ++t) acc[t] = (v8f){};

  for (int k0 = 0; k0 < K; k0 += BK) {
    const int kc = (K - k0 < BK) ? (K - k0) : BK;
    if (wave == 0) {
      // A chunk: rows [row0,row0+128) x cols [k0,k0+kc); OOB rows -> 0
      tdm_load_2d(0u, A + (size_t)row0 * lda + k0,
                  (u32)(lda - k0), (u32)(N_NODES - row0), (u32)kc, 128u, (u32)lda);
      // W chunk: rows [k0,k0+kc) x 128 cols
      tdm_load_2d((u32)(128 * BK * 4), W + (size_t)k0 * HID,
                  (u32)HID, (u32)(K - k0), (u32)HID, (u32)kc, (u32)HID);
      __builtin_amdgcn_s_wait_tensorcnt(0);
    }
    __syncthreads();
    // The TDM engine wrote smem behind the compiler's back; escape the LDS
    // pointers through an asm with a memory clobber so the sA/sW loads below
    // are not folded to undef (nothing in the IR visibly stores smem).
    asm volatile("" : : "r"(sA), "r"(sW) : "memory");
    for (int kk = 0; kk < kc; kk += 4) {
      v2f a;
      a.x = sA[lrow * kc + kk + khalf];
      a.y = sA[lrow * kc + kk + khalf + 1];
#pragma unroll
      for (int t = 0; t < 8; ++t) {
        v2f b;
        b.x = sW[(kk + khalf) * HID + t * 16 + ncb];
        b.y = sW[(kk + khalf + 1) * HID + t * 16 + ncb];
        acc[t] = __builtin_amdgcn_wmma_f32_16x16x4_f32(false, a, false, b,
                                                       (short)0, acc[t], false, false);
      }
    }
    __syncthreads();
  }

  if (tm < nRowTiles) {                          // wave-uniform guard
    const int rbase = (lane >> 4) * 8;
#pragma unroll
    for (int t = 0; t < 8; ++t)
#pragma unroll
      for (int r = 0; r < 8; ++r)
        xt[(size_t)(tm * 16 + rbase + r) * HID + t * 16 + ncb] = acc[t][r];
  }
}

// a_src[n] = xt[n,:]·att_src ; a_dst[n] = xt[n,:]·att_dst   (one wave / node)
__global__ __launch_bounds__(256) void node_att_kernel(const float* __restrict__ xt,
                                                       const float* __restrict__ att_src,
                                                       const float* __restrict__ att_dst,
                                                       float* __restrict__ a_src,
                                                       float* __restrict__ a_dst) {
  const int lane = threadIdx.x & 31;
  const int node = blockIdx.x * (blockDim.x >> 5) + (threadIdx.x >> 5);
  if (node >= N_NODES) return;
  const int c0 = lane * 4;
  float4 v  = *(const float4*)(xt + (size_t)node * HID + c0);
  float4 as = *(const float4*)(att_src + c0);
  float4 ad = *(const float4*)(att_dst + c0);
  float ss = v.x * as.x + v.y * as.y + v.z * as.z + v.w * as.w;
  float sd = v.x * ad.x + v.y * ad.y + v.z * ad.z + v.w * ad.w;
  ss = waveReduceSum(ss);
  sd = waveReduceSum(sd);
  if (lane == 0) { a_src[node] = ss; a_dst[node] = sd; }
}

__global__ void fill_kernel(float* __restrict__ p, int n, float v) {
  int i = blockIdx.x * blockDim.x + threadIdx.x;
  if (i < n) p[i] = v;
}

// sum of edge_attr columns (for self-loop mean feature)
__global__ __launch_bounds__(256) void edge_mean_accum(const float* __restrict__ edge_attr,
                                                       float* __restrict__ mean4) {
  const int lane = threadIdx.x & 31;
  int i = blockIdx.x * blockDim.x + threadIdx.x;
  float4 s = {0.f, 0.f, 0.f, 0.f};
  for (int e = i; e < N_EDGES; e += gridDim.x * blockDim.x) {
    float4 a = *(const float4*)(edge_attr + (size_t)e * 4);
    s.x += a.x; s.y += a.y; s.z += a.z; s.w += a.w;
  }
  s.x = waveReduceSum(s.x); s.y = waveReduceSum(s.y);
  s.z = waveReduceSum(s.z); s.w = waveReduceSum(s.w);
  if (lane == 0) {
    atomicAdd(mean4 + 0, s.x); atomicAdd(mean4 + 1, s.y);
    atomicAdd(mean4 + 2, s.z); atomicAdd(mean4 + 3, s.w);
  }
}

// scal[0..3] = W_e @ att_e (4-vector);  scal[4] = mean_ea · scal[0..3]
__global__ void prep_scal_kernel(const float* __restrict__ W_e, const float* __restrict__ att_e,
                                 const float* __restrict__ mean4, float* __restrict__ scal) {
  if (threadIdx.x != 0 || blockIdx.x != 0) return;
  float ael = 0.f;
  for (int c = 0; c < EDGE_DIM; ++c) {
    float s = 0.f;
    for (int j = 0; j < HID; ++j) s += W_e[c * HID + j] * att_e[j];
    scal[c] = s;
    ael += (mean4[c] * (1.0f / N_EDGES)) * s;
  }
  scal[4] = ael;
}

__global__ void edge_logits_kernel(const int* __restrict__ ei, const float* __restrict__ edge_attr,
                                   const float* __restrict__ a_src, const float* __restrict__ a_dst,
                                   const float* __restrict__ scal,
                                   float* __restrict__ alpha, float* __restrict__ mx) {
  int e = blockIdx.x * blockDim.x + threadIdx.x;
  if (e >= N_EDGES_T) return;
  int s, d; float ae;
  if (e < N_EDGES) {
    s = ei[e]; d = ei[N_EDGES + e];
    float4 a = *(const float4*)(edge_attr + (size_t)e * 4);
    ae = a.x * scal[0] + a.y * scal[1] + a.z * scal[2] + a.w * scal[3];
  } else {
    s = d = e - N_EDGES;
    ae = scal[4];
  }
  float al = a_src[s] + a_dst[d] + ae;
  al = (al >= 0.f) ? al : al * NEG_SLOPE;     // leaky_relu
  alpha[e] = al;
  atomicMaxF(mx + d, al);
}

__global__ void edge_exp_kernel(const int* __restrict__ ei, float* __restrict__ alpha,
                                const float* __restrict__ mx, float* __restrict__ denom) {
  int e = blockIdx.x * blockDim.x + threadIdx.x;
  if (e >= N_EDGES_T) return;
  int d = (e < N_EDGES) ? ei[N_EDGES + e] : (e - N_EDGES);
  float ex = expf(alpha[e] - mx[d]);
  alpha[e] = ex;                               // in-place: alpha now holds exp()
  atomicAdd(denom + d, ex);
}

// one wave per edge; 4 channels per lane; gather xt[src], scatter-add to agg[dst]
__global__ __launch_bounds__(256) void edge_agg_kernel(const int* __restrict__ ei,
                                                       const float* __restrict__ ex,
                                                       const float* __restrict__ denom,
                                                       const float* __restrict__ xt,
                                                       float* __restrict__ agg) {
  const int lane = threadIdx.x & 31;
  int e = blockIdx.x * (blockDim.x >> 5) + (threadIdx.x >> 5);
  if (e >= N_EDGES_T) return;
  int s, d;
  if (e < N_EDGES) { s = ei[e]; d = ei[N_EDGES + e]; }
  else             { s = d = e - N_EDGES; }
  float w = ex[e] / (denom[d] + 1e-16f);
  const int c0 = lane * 4;
  float4 v = *(const float4*)(xt + (size_t)s * HID + c0);
  float* dst = agg + (size_t)d * HID + c0;
  atomicAdd(dst + 0, v.x * w);
  atomicAdd(dst + 1, v.y * w);
  atomicAdd(dst + 2, v.z * w);
  atomicAdd(dst + 3, v.w * w);
}

// h = relu(layernorm(agg + b))   (one wave / node)
__global__ __launch_bounds__(256) void post_ln_kernel(const float* __restrict__ agg,
                                                      const float* __restrict__ bias,
                                                      const float* __restrict__ g,
                                                      const float* __restrict__ bln,
                                                      float* __restrict__ hout) {
  const int lane = threadIdx.x & 31;
  const int node = blockIdx.x * (blockDim.x >> 5) + (threadIdx.x >> 5);
  if (node >= N_NODES) return;
  const int c0 = lane * 4;
  float4 v = *(const float4*)(agg + (size_t)node * HID + c0);
  float4 b = *(const float4*)(bias + c0);
  v.x += b.x; v.y += b.y; v.z += b.z; v.w += b.w;
  float mu = waveReduceSum(v.x + v.y + v.z + v.w) * (1.f / HID);
  float4 dv = {v.x - mu, v.y - mu, v.z - mu, v.w - mu};
  float var = waveReduceSum(dv.x * dv.x + dv.y * dv.y + dv.z * dv.z + dv.w * dv.w) * (1.f / HID);
  float inv = rsqrtf(var + LN_EPS);
  float4 gg = *(const float4*)(g + c0);
  float4 bb = *(const float4*)(bln + c0);
  float4 o;
  o.x = fmaxf(dv.x * inv * gg.x + bb.x, 0.f);
  o.y = fmaxf(dv.y * inv * gg.y + bb.y, 0.f);
  o.z = fmaxf(dv.z * inv * gg.z + bb.z, 0.f);
  o.w = fmaxf(dv.w * inv * gg.w + bb.w, 0.f);
  *(float4*)(hout + (size_t)node * HID + c0) = o;
}

__global__ __launch_bounds__(256) void pool_kernel(const float* __restrict__ h,
                                                   const int* __restrict__ batch,
                                                   float* __restrict__ sums,
                                                   float* __restrict__ cnts) {
  const int lane = threadIdx.x & 31;
  const int node = blockIdx.x * (blockDim.x >> 5) + (threadIdx.x >> 5);
  if (node >= N_NODES) return;
  int g = batch[node];
  const int c0 = lane * 4;
  float4 v = *(const float4*)(h + (size_t)node * HID + c0);
  float* dst = sums + (size_t)g * HID + c0;
  atomicAdd(dst + 0, v.x); atomicAdd(dst + 1, v.y);
  atomicAdd(dst + 2, v.z); atomicAdd(dst + 3, v.w);
  if (lane == 0) atomicAdd(cnts + g, 1.0f);
}

__global__ void hg_kernel(const float* __restrict__ sums, const float* __restrict__ cnts,
                          float* __restrict__ hg) {
  int i = blockIdx.x * blockDim.x + threadIdx.x;
  if (i >= BATCHES * HID) return;
  hg[i] = sums[i] / cnts[i / HID];
}

__global__ void mlp1_kernel(const float* __restrict__ hg, const float* __restrict__ w1,
                            const float* __restrict__ b1, float* __restrict__ h1) {
  int i = blockIdx.x * blockDim.x + threadIdx.x;
  if (i >= BATCHES * HID) return;
  int g = i / HID, c = i % HID;
  float s = b1[c];
  for (int k = 0; k < HID; ++k) s += hg[g * HID + k] * w1[k * HID + c];
  h1[i] = fmaxf(s, 0.f);
}

__global__ void mlp2_kernel(const float* __restrict__ h1, const float* __restrict__ w2,
                            const float* __restrict__ b2, float* __restrict__ o) {
  int i = blockIdx.x * blockDim.x + threadIdx.x;
  if (i >= BATCHES * OUT_DIM) return;
  int g = i / OUT_DIM, c = i % OUT_DIM;
  float s = b2[c];
  for (int k = 0; k < HID; ++k) s += h1[g * HID + k] * w2[k * OUT_DIM + c];
  o[i] = s;
}

__global__ __launch_bounds__(256) void out_ln_kernel(const float* __restrict__ o,
                                                     const float* __restrict__ g,
                                                     const float* __restrict__ b,
                                                     float* __restrict__ out) {
  __shared__ float sm[OUT_DIM];
  const int row = blockIdx.x, c = threadIdx.x;
  float v = o[(size_t)row * OUT_DIM + c];
  sm[c] = v; __syncthreads();
  for (int s2 = OUT_DIM / 2; s2 > 0; s2 >>= 1) {
    if (c < s2) sm[c] += sm[c + s2];
    __syncthreads();
  }
  float mu = sm[0] * (1.f / OUT_DIM);
  __syncthreads();
  float dv = v - mu;
  sm[c] = dv * dv; __syncthreads();
  for (int s2 = OUT_DIM / 2; s2 > 0; s2 >>= 1) {
    if (c < s2) sm[c] += sm[c + s2];
    __syncthreads();
  }
  float inv = rsqrtf(sm[0] * (1.f / OUT_DIM) + LN_EPS);
  out[(size_t)row * OUT_DIM + c] = dv * inv * g[c] + b[c];
}

static inline int cdiv(int a, int b) { return (a + b - 1) / b; }

extern "C" void kernel_launch(void* const* d_in, const int* in_sizes, int n_in,
                              void* d_out, int out_size, void* d_ws, size_t ws_size,
                              hipStream_t stream) {
  // inputs: x, edge_index, edge_attr, batch, then params leaves in jax tree
  // order (dicts sorted by key): per layer: W, W_e, att_dst, att_e, att_src,
  // b, ln_b, ln_g ; then op_b1, op_b2, op_ln_b, op_ln_g, op_w1, op_w2
  const float* x         = (const float*)d_in[0];
  const int*   ei        = (const int*)d_in[1];
  const float* edge_attr = (const float*)d_in[2];
  const int*   batch     = (const int*)d_in[3];
  const float *W[3], *W_e[3], *att_dst[3], *att_e[3], *att_src[3], *bias[3], *ln_b[3], *ln_g[3];
  for (int p = 0; p < 3; ++p) {
    int base = 4 + 8 * p;
    W[p]       = (const float*)d_in[base + 0];
    W_e[p]     = (const float*)d_in[base + 1];
    att_dst[p] = (const float*)d_in[base + 2];
    att_e[p]   = (const float*)d_in[base + 3];
    att_src[p] = (const float*)d_in[base + 4];
    bias[p]    = (const float*)d_in[base + 5];
    ln_b[p]    = (const float*)d_in[base + 6];
    ln_g[p]    = (const float*)d_in[base + 7];
  }
  const float* op_b1   = (const float*)d_in[28];
  const float* op_b2   = (const float*)d_in[29];
  const float* op_ln_b = (const float*)d_in[30];
  const float* op_ln_g = (const float*)d_in[31];
  const float* op_w1   = (const float*)d_in[32];
  const float* op_w2   = (const float*)d_in[33];

  float* ws = (float*)d_ws;
  float* xt    = ws; ws += (size_t)N_NODES * HID;
  float* agg   = ws; ws += (size_t)N_NODES * HID;
  float* hbuf  = ws; ws += (size_t)N_NODES * HID;
  float* a_src = ws; ws += N_NODES;
  float* a_dst = ws; ws += N_NODES;
  float* alpha = ws; ws += N_EDGES_T;
  float* mx    = ws; ws += N_NODES;
  float* denom = ws; ws += N_NODES;
  float* mean4 = ws; ws += 4;
  float* scal  = ws; ws += 8;
  float* sums  = ws; ws += BATCHES * HID;
  float* cnts  = ws; ws += BATCHES;
  float* hg    = ws; ws += BATCHES * HID;
  float* h1    = ws; ws += BATCHES * HID;
  float* obuf  = ws; ws += BATCHES * OUT_DIM;

  // mean edge feature (layer-independent)
  fill_kernel<<<1, 32, 0, stream>>>(mean4, 4, 0.f);
  edge_mean_accum<<<512, 256, 0, stream>>>(edge_attr, mean4);

  const float* hin = x;
  int K = NODE_DIM;
  const int nRowTiles = N_NODES / 16;                      // 3125
  for (int p = 0; p < 3; ++p) {
    gemm_xt_wmma<<<cdiv(nRowTiles, 8), 256, 0, stream>>>(hin, K, K, W[p], xt);
    node_att_kernel<<<cdiv(N_NODES, 8), 256, 0, stream>>>(xt, att_src[p], att_dst[p], a_src, a_dst);
    prep_scal_kernel<<<1, 32, 0, stream>>>(W_e[p], att_e[p], mean4, scal);
    fill_kernel<<<cdiv(N_NODES * HID, 256), 256, 0, stream>>>(agg, N_NODES * HID, 0.f);
    fill_kernel<<<cdiv(N_NODES, 256), 256, 0, stream>>>(denom, N_NODES, 0.f);
    fill_kernel<<<cdiv(N_NODES, 256), 256, 0, stream>>>(mx, N_NODES, -3.0e38f);
    edge_logits_kernel<<<cdiv(N_EDGES_T, 256), 256, 0, stream>>>(ei, edge_attr, a_src, a_dst,
                                                                 scal, alpha, mx);
    edge_exp_kernel<<<cdiv(N_EDGES_T, 256), 256, 0, stream>>>(ei, alpha, mx, denom);
    edge_agg_kernel<<<cdiv(N_EDGES_T, 8), 256, 0, stream>>>(ei, alpha, denom, xt, agg);
    post_ln_kernel<<<cdiv(N_NODES, 8), 256, 0, stream>>>(agg, bias[p], ln_g[p], ln_b[p], hbuf);
    hin = hbuf;
    K = HID;
  }

  fill_kernel<<<cdiv(BATCHES * HID, 256), 256, 0, stream>>>(sums, BATCHES * HID, 0.f);
  fill_kernel<<<1, 64, 0, stream>>>(cnts, BATCHES, 0.f);
  pool_kernel<<<cdiv(N_NODES, 8), 256, 0, stream>>>(hbuf, batch, sums, cnts);
  hg_kernel<<<cdiv(BATCHES * HID, 256), 256, 0, stream>>>(sums, cnts, hg);
  mlp1_kernel<<<cdiv(BATCHES * HID, 256), 256, 0, stream>>>(hg, op_w1, op_b1, h1);
  mlp2_kernel<<<cdiv(BATCHES * OUT_DIM, 256), 256, 0, stream>>>(h1, op_w2, op_b2, obuf);
  out_ln_kernel<<<BATCHES, OUT_DIM, 0, stream>>>(obuf, op_ln_g, op_ln_b, (float*)d_out);
}